// ACMIL_GA_84705345012001
// MI455X (gfx1250) — compile-verified
//
#include <hip/hip_runtime.h>
#include <math.h>

typedef __attribute__((ext_vector_type(2))) float v2f;
typedef __attribute__((ext_vector_type(8))) float v8f;

// Problem constants
#define BB    8
#define NN    8192
#define DFEAT 1024
#define DINN  512
#define DATT  128
#define KK    5
#define NCLS  2
#define MROWS (BB * NN)        // 65536
#define NEGV  (-1000000000.0f)
#define NDROP 6

// ---------------------------------------------------------------------------
// Kernel 1: h = relu(x @ W_dr + b_dr)   (65536x1024)x(1024x512)
// wave tile: 16 rows x 64 cols, f32 WMMA 16x16x4, K-loop of 256 steps
// grid: 4096 blocks x 256 thr (8 waves, wave w -> n0 = w*64)
// ---------------------------------------------------------------------------
__global__ __launch_bounds__(256) void k_gemm_h(const float* __restrict__ X,
                                                const float* __restrict__ W,
                                                const float* __restrict__ bias,
                                                float* __restrict__ H) {
  const int lane = threadIdx.x & 31;
  const int wave = threadIdx.x >> 5;
  const int l15  = lane & 15;
  const int hi   = lane >> 4;            // 0 or 1
  const int m0   = blockIdx.x * 16;
  const int n0   = wave * 64;
  const int m    = m0 + l15;

  const float* xrow = X + (size_t)m * DFEAT + hi * 2;
  v8f acc0 = {}, acc1 = {}, acc2 = {}, acc3 = {};

  for (int k0 = 0; k0 < DFEAT; k0 += 4) {
    v2f a;
    a.x = xrow[k0];
    a.y = xrow[k0 + 1];
    const float* wr0 = W + (size_t)(k0 + hi * 2) * DINN;
    const float* wr1 = wr0 + DINN;
    v2f b0, b1, b2, b3;
    b0.x = wr0[n0 + l15];       b0.y = wr1[n0 + l15];
    b1.x = wr0[n0 + 16 + l15];  b1.y = wr1[n0 + 16 + l15];
    b2.x = wr0[n0 + 32 + l15];  b2.y = wr1[n0 + 32 + l15];
    b3.x = wr0[n0 + 48 + l15];  b3.y = wr1[n0 + 48 + l15];
    acc0 = __builtin_amdgcn_wmma_f32_16x16x4_f32(false, a, false, b0, (short)0, acc0, false, false);
    acc1 = __builtin_amdgcn_wmma_f32_16x16x4_f32(false, a, false, b1, (short)0, acc1, false, false);
    acc2 = __builtin_amdgcn_wmma_f32_16x16x4_f32(false, a, false, b2, (short)0, acc2, false, false);
    acc3 = __builtin_amdgcn_wmma_f32_16x16x4_f32(false, a, false, b3, (short)0, acc3, false, false);
  }

  const int col = n0 + l15;
  const float bi0 = bias[col], bi1 = bias[col + 16], bi2 = bias[col + 32], bi3 = bias[col + 48];
  const int rowBase = m0 + hi * 8;
#pragma unroll
  for (int i = 0; i < 8; i++) {
    float* hr = H + (size_t)(rowBase + i) * DINN + col;
    float v;
    v = acc0[i] + bi0; hr[0]  = v > 0.0f ? v : 0.0f;
    v = acc1[i] + bi1; hr[16] = v > 0.0f ? v : 0.0f;
    v = acc2[i] + bi2; hr[32] = v > 0.0f ? v : 0.0f;
    v = acc3[i] + bi3; hr[48] = v > 0.0f ? v : 0.0f;
  }
}

// ---------------------------------------------------------------------------
// Kernel 2: gate = tanh(h@W_v + b_v) * sigmoid(h@W_u + b_u)  -> (65536x128)
// wave tile 16x64; wave w: mTile = blk*4 + (w>>1), n0 = (w&1)*64. 1024 blocks.
// ---------------------------------------------------------------------------
__global__ __launch_bounds__(256) void k_gemm_gate(const float* __restrict__ H,
                                                   const float* __restrict__ Wv,
                                                   const float* __restrict__ bv,
                                                   const float* __restrict__ Wu,
                                                   const float* __restrict__ bu,
                                                   float* __restrict__ G) {
  const int lane = threadIdx.x & 31;
  const int wave = threadIdx.x >> 5;
  const int l15  = lane & 15;
  const int hi   = lane >> 4;
  const int m0   = (blockIdx.x * 4 + (wave >> 1)) * 16;
  const int n0   = (wave & 1) * 64;
  const int m    = m0 + l15;

  const float* hrow = H + (size_t)m * DINN + hi * 2;
  v8f av0 = {}, av1 = {}, av2 = {}, av3 = {};
  v8f au0 = {}, au1 = {}, au2 = {}, au3 = {};

  for (int k0 = 0; k0 < DINN; k0 += 4) {
    v2f a;
    a.x = hrow[k0];
    a.y = hrow[k0 + 1];
    const float* vr0 = Wv + (size_t)(k0 + hi * 2) * DATT;
    const float* vr1 = vr0 + DATT;
    const float* ur0 = Wu + (size_t)(k0 + hi * 2) * DATT;
    const float* ur1 = ur0 + DATT;
    v2f b0, b1, b2, b3, c0, c1, c2, c3;
    b0.x = vr0[n0 + l15];       b0.y = vr1[n0 + l15];
    b1.x = vr0[n0 + 16 + l15];  b1.y = vr1[n0 + 16 + l15];
    b2.x = vr0[n0 + 32 + l15];  b2.y = vr1[n0 + 32 + l15];
    b3.x = vr0[n0 + 48 + l15];  b3.y = vr1[n0 + 48 + l15];
    c0.x = ur0[n0 + l15];       c0.y = ur1[n0 + l15];
    c1.x = ur0[n0 + 16 + l15];  c1.y = ur1[n0 + 16 + l15];
    c2.x = ur0[n0 + 32 + l15];  c2.y = ur1[n0 + 32 + l15];
    c3.x = ur0[n0 + 48 + l15];  c3.y = ur1[n0 + 48 + l15];
    av0 = __builtin_amdgcn_wmma_f32_16x16x4_f32(false, a, false, b0, (short)0, av0, false, false);
    av1 = __builtin_amdgcn_wmma_f32_16x16x4_f32(false, a, false, b1, (short)0, av1, false, false);
    av2 = __builtin_amdgcn_wmma_f32_16x16x4_f32(false, a, false, b2, (short)0, av2, false, false);
    av3 = __builtin_amdgcn_wmma_f32_16x16x4_f32(false, a, false, b3, (short)0, av3, false, false);
    au0 = __builtin_amdgcn_wmma_f32_16x16x4_f32(false, a, false, c0, (short)0, au0, false, false);
    au1 = __builtin_amdgcn_wmma_f32_16x16x4_f32(false, a, false, c1, (short)0, au1, false, false);
    au2 = __builtin_amdgcn_wmma_f32_16x16x4_f32(false, a, false, c2, (short)0, au2, false, false);
    au3 = __builtin_amdgcn_wmma_f32_16x16x4_f32(false, a, false, c3, (short)0, au3, false, false);
  }

  const int col = n0 + l15;
  const float bv0 = bv[col], bv1 = bv[col + 16], bv2 = bv[col + 32], bv3 = bv[col + 48];
  const float bu0 = bu[col], bu1 = bu[col + 16], bu2 = bu[col + 32], bu3 = bu[col + 48];
  const int rowBase = m0 + hi * 8;
#pragma unroll
  for (int i = 0; i < 8; i++) {
    float* gr = G + (size_t)(rowBase + i) * DATT + col;
    float tv, su;
    tv = tanhf(av0[i] + bv0); su = 1.0f / (1.0f + expf(-(au0[i] + bu0))); gr[0]  = tv * su;
    tv = tanhf(av1[i] + bv1); su = 1.0f / (1.0f + expf(-(au1[i] + bu1))); gr[16] = tv * su;
    tv = tanhf(av2[i] + bv2); su = 1.0f / (1.0f + expf(-(au2[i] + bu2))); gr[32] = tv * su;
    tv = tanhf(av3[i] + bv3); su = 1.0f / (1.0f + expf(-(au3[i] + bu3))); gr[48] = tv * su;
  }
}

// ---------------------------------------------------------------------------
// Kernel 3: A = gate @ W_a + b_a   (65536x128)x(128x5). Wave per row.
// ---------------------------------------------------------------------------
__global__ __launch_bounds__(256) void k_attn(const float* __restrict__ G,
                                              const float* __restrict__ Wa,
                                              const float* __restrict__ ba,
                                              float* __restrict__ A) {
  const int lane = threadIdx.x & 31;
  const int wave = threadIdx.x >> 5;
  const int row  = blockIdx.x * 8 + wave;
  const float* g = G + (size_t)row * DATT;
  const float g0 = g[lane], g1 = g[lane + 32], g2 = g[lane + 64], g3 = g[lane + 96];
#pragma unroll
  for (int c = 0; c < KK; c++) {
    float acc = g0 * Wa[lane * KK + c] + g1 * Wa[(lane + 32) * KK + c] +
                g2 * Wa[(lane + 64) * KK + c] + g3 * Wa[(lane + 96) * KK + c];
#pragma unroll
    for (int off = 16; off > 0; off >>= 1) acc += __shfl_xor(acc, off, 32);
    if (lane == 0) A[(size_t)row * KK + c] = acc + ba[c];
  }
}

// ---------------------------------------------------------------------------
// Kernel 4: per (b,k): top-10 (value desc, index asc like lax.top_k), then set
// A[b, sel[rand_idx[i]], k] = NEG for the 6 chosen ranks. 40 blocks.
// ---------------------------------------------------------------------------
__global__ __launch_bounds__(256) void k_topk_mask(float* __restrict__ A,
                                                   const int* __restrict__ rand_idx) {
  const int b = blockIdx.x / KK;
  const int k = blockIdx.x % KK;
  float* Ab = A + (size_t)b * NN * KK + k;
  const int tid = threadIdx.x;

  float vals[NN / 256];
#pragma unroll
  for (int j = 0; j < NN / 256; j++) vals[j] = Ab[(size_t)(j * 256 + tid) * KK];

  __shared__ float sv[256];
  __shared__ int   si[256];
  __shared__ int   sel[10];

  for (int r = 0; r < 10; r++) {
    float bv = -3.402823466e38f;
    int   bi = 0x7fffffff;
#pragma unroll
    for (int j = 0; j < NN / 256; j++) {
      if (vals[j] > bv) { bv = vals[j]; bi = j * 256 + tid; }
    }
    sv[tid] = bv; si[tid] = bi;
    __syncthreads();
    for (int s = 128; s > 0; s >>= 1) {
      if (tid < s) {
        if (sv[tid + s] > sv[tid] || (sv[tid + s] == sv[tid] && si[tid + s] < si[tid])) {
          sv[tid] = sv[tid + s]; si[tid] = si[tid + s];
        }
      }
      __syncthreads();
    }
    if (tid == 0) sel[r] = si[0];
    __syncthreads();
    const int w = sel[r];
    if ((w & 255) == tid) vals[w >> 8] = -3.402823466e38f;  // remove from candidates
  }
  __syncthreads();
  if (tid < NDROP) {
    const int pos = sel[rand_idx[tid]];
    Ab[(size_t)pos * KK] = NEGV;
  }
}

// ---------------------------------------------------------------------------
// Kernel 5: softmax over N per (b,k); also writes masked A to d_out[96..].
// ---------------------------------------------------------------------------
__global__ __launch_bounds__(256) void k_softmax(const float* __restrict__ A,
                                                 float* __restrict__ S,
                                                 float* __restrict__ outA) {
  const int b = blockIdx.x / KK;
  const int k = blockIdx.x % KK;
  const float* Ab = A + (size_t)b * NN * KK + k;
  const int tid = threadIdx.x;
  __shared__ float red[256];

  float lmax = -3.402823466e38f;
  for (int j = 0; j < NN / 256; j++) {
    float v = Ab[(size_t)(j * 256 + tid) * KK];
    lmax = v > lmax ? v : lmax;
  }
  red[tid] = lmax;
  __syncthreads();
  for (int s = 128; s > 0; s >>= 1) {
    if (tid < s) red[tid] = red[tid + s] > red[tid] ? red[tid + s] : red[tid];
    __syncthreads();
  }
  const float mx = red[0];
  __syncthreads();

  float lsum = 0.0f;
  for (int j = 0; j < NN / 256; j++) {
    lsum += expf(Ab[(size_t)(j * 256 + tid) * KK] - mx);
  }
  red[tid] = lsum;
  __syncthreads();
  for (int s = 128; s > 0; s >>= 1) {
    if (tid < s) red[tid] += red[tid + s];
    __syncthreads();
  }
  const float rinv = 1.0f / red[0];

  for (int j = 0; j < NN / 256; j++) {
    const int n = j * 256 + tid;
    const size_t idx = ((size_t)b * NN + n) * KK + k;
    const float a = Ab[(size_t)n * KK];
    outA[idx] = a;                       // masked A output (1,B,N,K)
    S[idx]    = expf(a - mx) * rinv;
  }
}

// ---------------------------------------------------------------------------
// Kernel 6: Sm[b,n] = mean_k S[b,n,k]
// ---------------------------------------------------------------------------
__global__ __launch_bounds__(256) void k_slide_mean(const float* __restrict__ S,
                                                    float* __restrict__ Sm) {
  const int i = blockIdx.x * 256 + threadIdx.x;  // < 65536
  const float* s = S + (size_t)i * KK;
  Sm[i] = 0.2f * (s[0] + s[1] + s[2] + s[3] + s[4]);
}

// ---------------------------------------------------------------------------
// Kernel 7: partials of afeat[b,j,d] = sum_n h[b,n,d]*S'[b,n,j] (j=5 -> Sm)
// grid (32 nchunks, 2 dchunks, 8 b); 256 thr = d lanes. Deterministic.
// ---------------------------------------------------------------------------
__global__ __launch_bounds__(256) void k_afeat_part(const float* __restrict__ H,
                                                    const float* __restrict__ S,
                                                    const float* __restrict__ Sm,
                                                    float* __restrict__ P) {
  const int nchunk = blockIdx.x;
  const int d      = blockIdx.y * 256 + threadIdx.x;
  const int b      = blockIdx.z;
  const float* hb  = H + (size_t)b * NN * DINN;
  float acc[6] = {0, 0, 0, 0, 0, 0};
  for (int t = 0; t < 256; t++) {
    const int n = nchunk * 256 + t;
    const float hv = hb[(size_t)n * DINN + d];
    const float* s = S + ((size_t)b * NN + n) * KK;
    acc[0] += hv * s[0];
    acc[1] += hv * s[1];
    acc[2] += hv * s[2];
    acc[3] += hv * s[3];
    acc[4] += hv * s[4];
    acc[5] += hv * Sm[b * NN + n];
  }
#pragma unroll
  for (int j = 0; j < 6; j++)
    P[(((size_t)b * 6 + j) * DINN + d) * 32 + nchunk] = acc[j];
}

// ---------------------------------------------------------------------------
// Kernel 8: reduce 32 partials -> afeat[b,j,d]; j==5 goes to slide_feat out.
// ---------------------------------------------------------------------------
__global__ __launch_bounds__(256) void k_afeat_red(const float* __restrict__ P,
                                                   float* __restrict__ AF,
                                                   float* __restrict__ outSF) {
  const int idx = blockIdx.x * 256 + threadIdx.x;  // < 8*6*512
  const int b = idx / (6 * DINN);
  const int rem = idx % (6 * DINN);
  const int j = rem / DINN;
  const int d = rem % DINN;
  const float* p = P + (size_t)idx * 32;
  float s = 0.0f;
#pragma unroll
  for (int c = 0; c < 32; c++) s += p[c];
  AF[idx] = s;
  if (j == 5) outSF[b * DINN + d] = s;
}

// ---------------------------------------------------------------------------
// Kernel 9: heads. t<80: branch_outputs (K,B,C); t in [80,96): slide_output.
// ---------------------------------------------------------------------------
__global__ __launch_bounds__(128) void k_heads(const float* __restrict__ AF,
                                               const float* __restrict__ Wcls,
                                               const float* __restrict__ bcls,
                                               const float* __restrict__ Wsl,
                                               const float* __restrict__ bsl,
                                               float* __restrict__ out) {
  const int t = threadIdx.x;
  if (t < 80) {
    const int k = t >> 4, b = (t >> 1) & 7, c = t & 1;
    const float* af = AF + ((size_t)b * 6 + k) * DINN;
    const float* w  = Wcls + (size_t)k * DINN * NCLS + c;
    float s = 0.0f;
    for (int d = 0; d < DINN; d++) s += af[d] * w[(size_t)d * NCLS];
    out[t] = s + bcls[k * NCLS + c];
  } else if (t < 96) {
    const int i = t - 80, b = i >> 1, c = i & 1;
    const float* sf = AF + ((size_t)b * 6 + 5) * DINN;
    float s = 0.0f;
    for (int d = 0; d < DINN; d++) s += sf[d] * Wsl[(size_t)d * NCLS + c];
    out[80 + i] = s + bsl[c];
  }
}

// ---------------------------------------------------------------------------
extern "C" void kernel_launch(void* const* d_in, const int* in_sizes, int n_in,
                              void* d_out, int out_size, void* d_ws, size_t ws_size,
                              hipStream_t stream) {
  const float* x      = (const float*)d_in[0];
  const float* W_dr   = (const float*)d_in[1];
  const float* b_dr   = (const float*)d_in[2];
  const float* W_v    = (const float*)d_in[3];
  const float* b_v    = (const float*)d_in[4];
  const float* W_u    = (const float*)d_in[5];
  const float* b_u    = (const float*)d_in[6];
  const float* W_a    = (const float*)d_in[7];
  const float* b_a    = (const float*)d_in[8];
  const float* W_cls  = (const float*)d_in[9];
  const float* b_cls  = (const float*)d_in[10];
  const float* W_sl   = (const float*)d_in[11];
  const float* b_sl   = (const float*)d_in[12];
  const int*   ridx   = (const int*)d_in[13];

  float* out = (float*)d_out;
  // Output layout: [0,80) branch, [80,96) slide_out, [96,327776) A, [327776,331872) slide_feat
  float* outA  = out + 96;
  float* outSF = out + 96 + (size_t)BB * NN * KK;

  float* ws   = (float*)d_ws;
  float* h    = ws;                              // 65536*512
  float* gate = h + (size_t)MROWS * DINN;        // 65536*128
  float* A    = gate + (size_t)MROWS * DATT;     // 65536*5
  float* S    = A + (size_t)MROWS * KK;          // 65536*5
  float* Sm   = S + (size_t)MROWS * KK;          // 65536
  float* P    = Sm + (size_t)MROWS;              // 8*6*512*32
  float* AF   = P + (size_t)BB * 6 * DINN * 32;  // 8*6*512

  k_gemm_h<<<dim3(MROWS / 16), dim3(256), 0, stream>>>(x, W_dr, b_dr, h);
  k_gemm_gate<<<dim3(MROWS / 64), dim3(256), 0, stream>>>(h, W_v, b_v, W_u, b_u, gate);
  k_attn<<<dim3(MROWS / 8), dim3(256), 0, stream>>>(gate, W_a, b_a, A);
  k_topk_mask<<<dim3(BB * KK), dim3(256), 0, stream>>>(A, ridx);
  k_softmax<<<dim3(BB * KK), dim3(256), 0, stream>>>(A, S, outA);
  k_slide_mean<<<dim3(MROWS / 256), dim3(256), 0, stream>>>(S, Sm);
  k_afeat_part<<<dim3(32, 2, BB), dim3(256), 0, stream>>>(h, S, Sm, P);
  k_afeat_red<<<dim3((BB * 6 * DINN) / 256), dim3(256), 0, stream>>>(P, AF, outSF);
  k_heads<<<dim3(1), dim3(128), 0, stream>>>(AF, W_cls, b_cls, W_sl, b_sl, out);
}